// SMPLNN_26534307954892
// MI455X (gfx1250) — compile-verified
//
#include <hip/hip_runtime.h>
#include <hip/hip_bf16.h>

#define N_PTS   100000
#define N_VERTS 6890
#define N_JOINTS 24

// Verts staged to LDS in 2 chunks of 3456 (= 216 tiles of 16). 6912 >= 6890,
// pad entries get |v|^2 = 3e38 so they never win the argmin.
#define CHUNK   3456
#define CSTRIDE 3488            // +32 dwords so xs/zs (and ys/ss) windows hit disjoint LDS banks
#define NCHUNK  2
#define TILES_PER_CHUNK (CHUNK / 16)

typedef __attribute__((ext_vector_type(2))) float v2f;
typedef __attribute__((ext_vector_type(8))) float v8f;

// ---------------------------------------------------------------------------
// Kernel 1: nearest-vertex search via V_WMMA_F32_16X16X4_F32.
// score(p, v) = -2*x.v + |v|^2   (|x|^2 dropped: constant per row).
// A (16x4 f32): lanes 0-15 hold K=0,1 = (-2x,-2y); lanes 16-31 hold K=2,3 = (-2z, 1).
// B (4x16 f32): V0 = row K0 (lanes 0-15) / K2 (lanes 16-31); V1 = K1 / K3.
// D (16x16 f32): VGPR r, lane L -> M = r + 8*(L>=16), N = L&15.
// Each wave scores 16 points against all verts; 8 waves / block -> 128 pts/block.
// ---------------------------------------------------------------------------
__global__ __launch_bounds__(256)
void smplnn_nn_kernel(const float* __restrict__ xyz,
                      const float* __restrict__ verts,
                      int* __restrict__ pidx) {
  __shared__ float smem[4 * CSTRIDE];
  float* xs  = smem;
  float* zs  = smem + CSTRIDE;      // bank offset 32 vs xs
  float* ys  = smem + 2 * CSTRIDE;  // bank offset 0
  float* ssq = smem + 3 * CSTRIDE;  // bank offset 32 vs ys

  const int tid  = threadIdx.x;
  const int lane = tid & 31;
  const int wave = tid >> 5;
  const int half = lane >> 4;   // 0: lanes 0-15, 1: lanes 16-31
  const int col  = lane & 15;

  // A fragment (loop invariant). Clamp point index so EXEC stays full for WMMA.
  const int p0 = blockIdx.x * 128 + wave * 16;
  int pl = p0 + col;
  if (pl > N_PTS - 1) pl = N_PTS - 1;
  v2f a;
  if (half == 0) { a.x = -2.0f * xyz[3 * pl + 0]; a.y = -2.0f * xyz[3 * pl + 1]; }
  else           { a.x = -2.0f * xyz[3 * pl + 2]; a.y = 1.0f; }

  float best[8];
  int   bidx[8];
#pragma unroll
  for (int r = 0; r < 8; ++r) { best[r] = 3.0e38f; bidx[r] = 0; }

  for (int c = 0; c < NCHUNK; ++c) {
    // Cooperative stage: (x, y, z, |v|^2) SoA into LDS.
    for (int i = tid; i < CHUNK; i += 256) {
      const int g = c * CHUNK + i;
      if (g < N_VERTS) {
        const float vx = verts[3 * g + 0];
        const float vy = verts[3 * g + 1];
        const float vz = verts[3 * g + 2];
        xs[i] = vx; ys[i] = vy; zs[i] = vz;
        ssq[i] = vx * vx + vy * vy + vz * vz;
      } else {
        xs[i] = 0.0f; ys[i] = 0.0f; zs[i] = 0.0f; ssq[i] = 3.0e38f;
      }
    }
    __syncthreads();

#pragma unroll 2
    for (int t = 0; t < TILES_PER_CHUNK; ++t) {
      const int nb = t * 16 + col;
      v2f b;
      if (half == 0) { b.x = xs[nb]; b.y = ys[nb]; }    // rows K0, K1
      else           { b.x = zs[nb]; b.y = ssq[nb]; }   // rows K2, K3
      v8f cz = {};
      // (neg_a, A, neg_b, B, c_mod, C, reuse_a, reuse_b)
      v8f d = __builtin_amdgcn_wmma_f32_16x16x4_f32(
          false, a, false, b, (short)0, cz, false, false);
      const int n = c * CHUNK + t * 16 + col;
#pragma unroll
      for (int r = 0; r < 8; ++r) {
        if (d[r] < best[r]) { best[r] = d[r]; bidx[r] = n; }  // strict <: keeps first min
      }
    }
    __syncthreads();
  }

  // Reduce across each 16-lane half (row M = r + 8*half lives in one half).
#pragma unroll
  for (int r = 0; r < 8; ++r) {
    float bv = best[r];
    int   bi = bidx[r];
#pragma unroll
    for (int m = 1; m < 16; m <<= 1) {
      const float ov = __shfl_xor(bv, m, 32);
      const int   oi = __shfl_xor(bi, m, 32);
      if (ov < bv || (ov == bv && oi < bi)) { bv = ov; bi = oi; }
    }
    if (col == r) {
      const int p = p0 + r + half * 8;
      if (p < N_PTS) pidx[p] = bi;
    }
  }
}

// ---------------------------------------------------------------------------
// Kernel 2: per-point blend of 24 bone transforms + point/rotation transform.
// ---------------------------------------------------------------------------
__global__ __launch_bounds__(256)
void smplnn_xform_kernel(const float* __restrict__ xyz,
                         const float* __restrict__ rot,
                         const float* __restrict__ sw,
                         const float* __restrict__ bones,
                         const int* __restrict__ pidx,
                         float* __restrict__ xbar,
                         float* __restrict__ rbar,
                         float* __restrict__ tfwd) {
  __shared__ float B[N_JOINTS * 16];
  for (int i = threadIdx.x; i < N_JOINTS * 16; i += 256) B[i] = bones[i];
  __syncthreads();

  const int p = blockIdx.x * 256 + threadIdx.x;
  if (p >= N_PTS) return;
  const int v = pidx[p];

  float T[16];
#pragma unroll
  for (int k = 0; k < 16; ++k) T[k] = 0.0f;
#pragma unroll 4
  for (int j = 0; j < N_JOINTS; ++j) {
    const float w = sw[v * N_JOINTS + j];
#pragma unroll
    for (int k = 0; k < 16; ++k) T[k] = fmaf(w, B[j * 16 + k], T[k]);
  }

  const float x = xyz[3 * p + 0], y = xyz[3 * p + 1], z = xyz[3 * p + 2];
#pragma unroll
  for (int r = 0; r < 3; ++r)
    xbar[3 * p + r] =
        fmaf(T[4 * r + 0], x, fmaf(T[4 * r + 1], y, fmaf(T[4 * r + 2], z, T[4 * r + 3])));

  const float q0 = rot[4 * p + 0], q1 = rot[4 * p + 1];
  const float q2 = rot[4 * p + 2], q3 = rot[4 * p + 3];
  const float inv = rsqrtf(q0 * q0 + q1 * q1 + q2 * q2 + q3 * q3);
  const float rr = q0 * inv, qx = q1 * inv, qy = q2 * inv, qz = q3 * inv;

  float R[9];
  R[0] = 1.0f - 2.0f * (qy * qy + qz * qz);
  R[1] = 2.0f * (qx * qy - rr * qz);
  R[2] = 2.0f * (qx * qz + rr * qy);
  R[3] = 2.0f * (qx * qy + rr * qz);
  R[4] = 1.0f - 2.0f * (qx * qx + qz * qz);
  R[5] = 2.0f * (qy * qz - rr * qx);
  R[6] = 2.0f * (qx * qz - rr * qy);
  R[7] = 2.0f * (qy * qz + rr * qx);
  R[8] = 1.0f - 2.0f * (qx * qx + qy * qy);

#pragma unroll
  for (int i = 0; i < 3; ++i)
#pragma unroll
    for (int j = 0; j < 3; ++j)
      rbar[9 * p + 3 * i + j] =
          fmaf(T[4 * i + 0], R[j],
               fmaf(T[4 * i + 1], R[3 + j], T[4 * i + 2] * R[6 + j]));

#pragma unroll
  for (int k = 0; k < 16; ++k) tfwd[16 * p + k] = T[k];
}

extern "C" void kernel_launch(void* const* d_in, const int* in_sizes, int n_in,
                              void* d_out, int out_size, void* d_ws, size_t ws_size,
                              hipStream_t stream) {
  const float* xyz   = (const float*)d_in[0];   // (100000, 3)
  const float* rot   = (const float*)d_in[1];   // (100000, 4)
  const float* verts = (const float*)d_in[2];   // (6890, 3)
  const float* sw    = (const float*)d_in[3];   // (6890, 24)
  const float* bones = (const float*)d_in[4];   // (24, 4, 4)
  float* out = (float*)d_out;                   // x_bar | rotation_bar | T_fwd
  int* pidx = (int*)d_ws;                       // 100000 ints of scratch

  const int nb1 = (N_PTS + 127) / 128;          // 128 points per block (8 waves x 16)
  smplnn_nn_kernel<<<nb1, 256, 0, stream>>>(xyz, verts, pidx);

  const int nb2 = (N_PTS + 255) / 256;
  smplnn_xform_kernel<<<nb2, 256, 0, stream>>>(
      xyz, rot, sw, bones, pidx,
      out,                  // x_bar:        100000*3
      out + 3 * N_PTS,      // rotation_bar: 100000*9
      out + 12 * N_PTS);    // T_fwd:        100000*16
}